// GNNModel_59115929862878
// MI455X (gfx1250) — compile-verified
//
#include <hip/hip_runtime.h>
#include <cstdint>

#define IN_CH 495
#define HCH   16

typedef __attribute__((ext_vector_type(2))) float v2f;
typedef __attribute__((ext_vector_type(8))) float v8f;

__device__ __forceinline__ v8f wmma_f32(v2f a, v2f b, v8f c) {
  return __builtin_amdgcn_wmma_f32_16x16x4_f32(false, a, false, b,
                                               (short)0, c, false, false);
}

// ---------------------------------------------------------------- utilities
__global__ void fill_f32_kernel(float* __restrict__ p, float v, int n) {
  int t = blockIdx.x * blockDim.x + threadIdx.x;
  if (t < n) p[t] = v;
}

// deg[dst] += 1 per edge (deg pre-filled with 1.0 for self-loops)
__global__ void deg_count_kernel(const int* __restrict__ edges, float* __restrict__ degf, int E) {
  int e = blockIdx.x * blockDim.x + threadIdx.x;
  if (e < E) atomicAdd(&degf[edges[E + e]], 1.0f);
}

// in-place dis[i] = rsqrt(deg[i])
__global__ void rsqrt_kernel(float* __restrict__ p, int n) {
  int t = blockIdx.x * blockDim.x + threadIdx.x;
  if (t < n) p[t] = rsqrtf(p[t]);
}

// ---------------------------------------------------------------- GEMM (WMMA)
// out[M x 16] = A[M x K] @ W[K x 16], fp32 via V_WMMA_F32_16X16X4_F32.
// One wave per 16-row tile; branch-free main K loop, unrolled x2 with two
// independent accumulators; single guarded tail chunk for K % 4.
__global__ void gemm16_wmma_kernel(const float* __restrict__ A,
                                   const float* __restrict__ W,
                                   float* __restrict__ out,
                                   int nTiles, int K) {
  int wave = (blockIdx.x * blockDim.x + threadIdx.x) >> 5;
  int lane = threadIdx.x & 31;
  if (wave >= nTiles) return;            // wave-uniform: EXEC stays all-1s

  int half = lane >> 4;                  // half 0: K=k0,k0+1   half 1: K=k0+2,k0+3
  int l    = lane & 15;                  // A: row-in-tile, B: output column
  const float* arow = A + (size_t)(wave * 16 + l) * (size_t)K;
  const float* wcol = W + l;

  v8f c0 = {}, c1 = {};
  int kFull = K >> 2;                    // full 4-wide chunks (no bounds checks)
  int kk = 0;
  for (; kk + 1 < kFull; kk += 2) {
    int ka = kk * 4 + half * 2;
    int kb = ka + 4;
    // speculative stream-ahead prefetch; invalid addresses silently dropped
    __builtin_prefetch(arow + ka + 40, 0, 1);
    v2f a0, b0, a1, b1;
    a0.x = arow[ka];            a0.y = arow[ka + 1];
    b0.x = wcol[ka * HCH];      b0.y = wcol[(ka + 1) * HCH];
    a1.x = arow[kb];            a1.y = arow[kb + 1];
    b1.x = wcol[kb * HCH];      b1.y = wcol[(kb + 1) * HCH];
    c0 = wmma_f32(a0, b0, c0);
    c1 = wmma_f32(a1, b1, c1);
  }
  for (; kk < kFull; ++kk) {             // at most one leftover full chunk
    int ka = kk * 4 + half * 2;
    v2f a, b;
    a.x = arow[ka];             a.y = arow[ka + 1];
    b.x = wcol[ka * HCH];       b.y = wcol[(ka + 1) * HCH];
    c0 = wmma_f32(a, b, c0);
  }
  if (K & 3) {                           // guarded tail chunk (runs once)
    int ka = kFull * 4 + half * 2;
    v2f a, b;
    a.x = (ka     < K) ? arow[ka]             : 0.0f;
    a.y = (ka + 1 < K) ? arow[ka + 1]         : 0.0f;
    b.x = (ka     < K) ? wcol[ka * HCH]       : 0.0f;
    b.y = (ka + 1 < K) ? wcol[(ka + 1) * HCH] : 0.0f;
    c0 = wmma_f32(a, b, c0);
  }

  // C layout: VGPR r, lanes 0-15 -> row r col l; lanes 16-31 -> row r+8 col l
  float* obase = out + ((size_t)wave * 16 + half * 8) * HCH + l;
#pragma unroll
  for (int r = 0; r < 8; ++r) obase[r * HCH] = c0[r] + c1[r];
}

// ---------------------------------------------------------------- edge scatter
// acc[dst,c] += tmp[src,c] * dis[src]*dis[dst].  channel -> lane: each edge is
// one contiguous 64B burst of global_atomic_add_f32 into L2-resident acc.
__global__ void scatter_kernel(const float* __restrict__ tmp,
                               const int* __restrict__ edges,
                               const float* __restrict__ dis,
                               float* __restrict__ acc, int E) {
  int t = blockIdx.x * blockDim.x + threadIdx.x;
  if (t >= E * HCH) return;
  int e = t >> 4, ch = t & 15;
  int s = edges[e];
  int d = edges[E + e];
  float nrm = dis[s] * dis[d];
  atomicAdd(&acc[(size_t)d * HCH + ch], tmp[(size_t)s * HCH + ch] * nrm);
}

// h[i,c] = relu(acc[i,c] + tmp[i,c]*dis[i]^2 + b[c])   (self-loop + bias + relu)
__global__ void combine_kernel(const float* __restrict__ acc,
                               const float* __restrict__ tmp,
                               const float* __restrict__ dis,
                               const float* __restrict__ bias,
                               float* __restrict__ hout, int N) {
  int t = blockIdx.x * blockDim.x + threadIdx.x;
  if (t >= N * HCH) return;
  int i = t >> 4, ch = t & 15;
  float ds = dis[i];
  float v = acc[t] + tmp[t] * ds * ds + bias[ch];
  hout[t] = v > 0.0f ? v : 0.0f;
}

// ---------------------------------------------------------------- edge head
// Per wave: 16 edges. A = [h[src] | h[dst]] (16x32) gathered per-lane,
// C = A @ Wf0 via 8 f32 WMMAs (two accumulators); bias+relu; dot with Wf1
// via xor-butterfly within each 16-lane half.
__global__ void edge_head_wmma_kernel(const float* __restrict__ h,
                                      const int* __restrict__ edges,
                                      const float* __restrict__ Wf0,
                                      const float* __restrict__ bf0,
                                      const float* __restrict__ Wf1,
                                      const float* __restrict__ bf1,
                                      float* __restrict__ out, int E) {
  int wave = (blockIdx.x * blockDim.x + threadIdx.x) >> 5;
  int lane = threadIdx.x & 31;
  if (wave >= (E >> 4)) return;          // wave-uniform

  int half = lane >> 4, l = lane & 15;
  int e = wave * 16 + l;                 // lane's edge (A-matrix row)
  int s = edges[e];
  int d = edges[E + e];
  const float* hs = h + (size_t)s * HCH;
  const float* hd = h + (size_t)d * HCH;

  v8f c0 = {}, c1 = {};
#pragma unroll
  for (int kk = 0; kk < 8; kk += 2) {
    int ka = kk * 4 + half * 2;          // even; never straddles the 16-split
    int kb = ka + 4;
    const float* fa = (ka < HCH) ? (hs + ka) : (hd + (ka - HCH));
    const float* fb = (kb < HCH) ? (hs + kb) : (hd + (kb - HCH));
    v2f a0, b0, a1, b1;
    a0.x = fa[0];  a0.y = fa[1];
    b0.x = Wf0[ka * HCH + l];  b0.y = Wf0[(ka + 1) * HCH + l];
    a1.x = fb[0];  a1.y = fb[1];
    b1.x = Wf0[kb * HCH + l];  b1.y = Wf0[(kb + 1) * HCH + l];
    c0 = wmma_f32(a0, b0, c0);
    c1 = wmma_f32(a1, b1, c1);
  }

  float bias0 = bf0[l];
  float w1    = Wf1[l];
  float res[8];
#pragma unroll
  for (int r = 0; r < 8; ++r) {
    float v = c0[r] + c1[r] + bias0;
    v = v > 0.0f ? v : 0.0f;
    v *= w1;
    v += __shfl_xor(v, 1, 32);           // reduce over 16 columns
    v += __shfl_xor(v, 2, 32);
    v += __shfl_xor(v, 4, 32);
    v += __shfl_xor(v, 8, 32);
    res[r] = v;
  }
  if (l == 0) {                          // lanes 0 (rows 0-7) and 16 (rows 8-15)
    float bb = bf1[0];
    float* o = out + (size_t)wave * 16 + half * 8;
#pragma unroll
    for (int r = 0; r < 8; ++r) o[r] = res[r] + bb;
  }
}

// ---------------------------------------------------------------- launch
extern "C" void kernel_launch(void* const* d_in, const int* in_sizes, int n_in,
                              void* d_out, int out_size, void* d_ws, size_t ws_size,
                              hipStream_t stream) {
  const float* x   = (const float*)d_in[0];
  const int* edges = (const int*)d_in[1];          // [2E]: src then dst
  const float* W0  = (const float*)d_in[2];
  const float* b0  = (const float*)d_in[3];
  const float* W1  = (const float*)d_in[4];
  const float* b1  = (const float*)d_in[5];
  const float* W2  = (const float*)d_in[6];
  const float* b2  = (const float*)d_in[7];
  const float* Wf0 = (const float*)d_in[8];
  const float* bf0 = (const float*)d_in[9];
  const float* Wf1 = (const float*)d_in[10];
  const float* bf1 = (const float*)d_in[11];
  float* outp = (float*)d_out;

  const int N = in_sizes[0] / IN_CH;               // 100000
  const int E = in_sizes[1] / 2;                   // 3200000

  // workspace: dis | tmp | acc | hA | hB
  float* dis = (float*)d_ws;
  float* tmp = dis + N;
  float* acc = tmp + (size_t)N * HCH;
  float* hA  = acc + (size_t)N * HCH;
  float* hB  = hA  + (size_t)N * HCH;

  const int BLK = 256;
  auto cdiv = [](long long a, long long b) { return (int)((a + b - 1) / b); };

  // ---- degrees & symmetric norm
  fill_f32_kernel<<<cdiv(N, BLK), BLK, 0, stream>>>(dis, 1.0f, N);         // self-loop
  deg_count_kernel<<<cdiv(E, BLK), BLK, 0, stream>>>(edges, dis, E);
  rsqrt_kernel<<<cdiv(N, BLK), BLK, 0, stream>>>(dis, N);

  const int nTiles  = N / 16;                      // 6250
  const int gemmBlk = cdiv((long long)nTiles * 32, BLK);
  const int nc      = N * HCH;

  const float* layerW[3] = {W0, W1, W2};
  const float* layerB[3] = {b0, b1, b2};
  const float* layerIn[3] = {x, hA, hB};
  float*       layerOut[3] = {hA, hB, hA};
  const int    layerK[3] = {IN_CH, HCH, HCH};

  for (int lyr = 0; lyr < 3; ++lyr) {
    fill_f32_kernel<<<cdiv(nc, BLK), BLK, 0, stream>>>(acc, 0.0f, nc);
    gemm16_wmma_kernel<<<gemmBlk, BLK, 0, stream>>>(layerIn[lyr], layerW[lyr],
                                                    tmp, nTiles, layerK[lyr]);
    scatter_kernel<<<cdiv((long long)E * HCH, BLK), BLK, 0, stream>>>(tmp, edges, dis, acc, E);
    combine_kernel<<<cdiv(nc, BLK), BLK, 0, stream>>>(acc, tmp, dis, layerB[lyr],
                                                      layerOut[lyr], N);
  }

  // ---- edge MLP head (reads hA = layer-2 output)
  const int headBlk = cdiv((long long)(E / 16) * 32, BLK);
  edge_head_wmma_kernel<<<headBlk, BLK, 0, stream>>>(hA, edges, Wf0, bf0, Wf1, bf1,
                                                     outp, E);
}